// PFD_13735305412709
// MI455X (gfx1250) — compile-verified
//
#include <hip/hip_runtime.h>

// MI455X / gfx1250. Memory-bound fused PFA+PVM kernel:
// one workgroup per batch, all tensors staged in LDS (CDNA5: 320KB/WGP),
// Gram matrices via V_WMMA_F32_16X16X4_F32, single-pass HBM traffic.

typedef __attribute__((ext_vector_type(2))) float v2f;
typedef __attribute__((ext_vector_type(8))) float v8f;

constexpr int TT  = 17;    // tokens
constexpr int DD  = 768;   // feature dim
constexpr int RS  = 772;   // padded LDS row stride (772 % 64 == 4 -> conflict-free wmma frag loads)
constexpr int NW  = 8;     // waves per block (wave32)
constexpr int BLK = 256;

__global__ __launch_bounds__(BLK) void pfd_fused_kernel(
    const float* __restrict__ mat,
    const float* __restrict__ mat1,
    const float* __restrict__ mat2,
    float* __restrict__ out)
{
    extern __shared__ float smem[];
    float* A   = smem;            // matrix   [TT][RS]
    float* P   = A  + TT * RS;    // pwf      [TT][RS]
    float* M2  = P  + TT * RS;    // matrix2  [TT][RS]
    float* SIM = M2 + TT * RS;    // raw dot grams [2][TT][32]
    float* NRM = SIM + 2 * TT * 32;          // norms [2][TT]
    int*   IND = (int*)(NRM + 2 * TT);       // argmax [2][TT]

    const int tid  = threadIdx.x;
    const int lane = tid & 31;
    const int wid  = tid >> 5;
    const int b    = blockIdx.x;

    const size_t base = (size_t)b * (TT * DD);
    const float4* gm  = (const float4*)(mat  + base);
    const float4* gm1 = (const float4*)(mat1 + base);
    const float4* gm2 = (const float4*)(mat2 + base);

    // ---- Phase 1: stage inputs in LDS, fuse pwf = matrix * matrix1 ----
    for (int i = tid; i < TT * (DD / 4); i += BLK) {
        int row = i / (DD / 4);
        int c4  = i - row * (DD / 4);
        float4 a = gm[i];
        float4 x = gm1[i];
        float4 c = gm2[i];
        float4 p = make_float4(a.x * x.x, a.y * x.y, a.z * x.z, a.w * x.w);
        int off = row * RS + c4 * 4;
        *(float4*)&A[off]  = a;
        *(float4*)&P[off]  = p;
        *(float4*)&M2[off] = c;
    }
    __syncthreads();

    // ---- Phase 2a: row norms of pwf and matrix2 (wave-parallel reductions) ----
    for (int r = wid; r < 2 * TT; r += NW) {
        const float* src = (r < TT) ? &P[r * RS] : &M2[(r - TT) * RS];
        float s = 0.f;
        for (int j = lane; j < DD; j += 32) { float v = src[j]; s += v * v; }
        #pragma unroll
        for (int off = 16; off > 0; off >>= 1) s += __shfl_xor(s, off, 32);
        if (lane == 0) NRM[r] = fmaxf(sqrtf(s), 1e-8f);
    }

    // ---- Phase 2b: Gram matrices via WMMA f32 16x16x4 ----
    // wave w: gram = w>>2 (0: matrix.pwf^T, 1: matrix.matrix2^T), tile (mi, ni)
    {
        const int which = wid >> 2;
        const int mi    = (wid >> 1) & 1;
        const int ni    = wid & 1;
        const float* Bsrc = which ? M2 : P;

        const int half = lane >> 4;     // K sub-pair select per ISA layout
        const int l    = lane & 15;
        int mrow = mi * 16 + l; if (mrow >= TT) mrow = TT - 1;  // clamp: padded
        int nrow = ni * 16 + l; if (nrow >= TT) nrow = TT - 1;  // entries never read
        const float* arow = &A[mrow * RS + 2 * half];
        const float* brow = &Bsrc[nrow * RS + 2 * half];

        v8f acc = {};
        #pragma unroll 4
        for (int k = 0; k < DD; k += 4) {
            v2f av = *(const v2f*)(arow + k);   // ds_load_b64, conflict-free
            v2f bv = *(const v2f*)(brow + k);
            acc = __builtin_amdgcn_wmma_f32_16x16x4_f32(
                      false, av, false, bv, (short)0, acc, false, false);
        }
        // spill valid accumulator entries to LDS (guards drop padded rows/cols)
        #pragma unroll
        for (int r = 0; r < 8; ++r) {
            int mg = mi * 16 + r + 8 * half;
            int ng = ni * 16 + l;
            if (mg < TT && ng < TT) SIM[(which * TT + mg) * 32 + ng] = acc[r];
        }
    }
    __syncthreads();

    // ---- Phase 3: per-row argmax (divide only by column norm; row norm is
    //      a positive constant per row so it cannot change the argmax) ----
    if (tid < 2 * TT) {
        int which = tid >= TT ? 1 : 0;
        int m = tid - which * TT;
        const float* srow = &SIM[(which * TT + m) * 32];
        const float* nrm  = &NRM[which * TT];
        float best = srow[0] / nrm[0];
        int bi = 0;
        for (int n = 1; n < TT; ++n) {
            float v = srow[n] / nrm[n];
            if (v > best) { best = v; bi = n; }   // first-max wins (jnp.argmax)
        }
        IND[tid] = bi;
    }
    __syncthreads();

    // ---- Phase 4: fused gather + add, single coalesced output pass ----
    float4* gout = (float4*)(out + (size_t)b * (2 * TT * DD));
    for (int i = tid; i < 2 * TT * (DD / 4); i += BLK) {
        int row = i / (DD / 4);
        int c4  = i - row * (DD / 4);
        float4 res;
        if (row < TT) {                       // PFA: matrix[m] + pwf[ind1[m]]
            int j = IND[row];
            float4 x = *(const float4*)&A[row * RS + c4 * 4];
            float4 p = *(const float4*)&P[j * RS + c4 * 4];
            res = make_float4(x.x + p.x, x.y + p.y, x.z + p.z, x.w + p.w);
        } else {                              // PVM: matrix[ind2[m]] + matrix2[m]
            int m = row - TT;
            int j = IND[TT + m];
            float4 x = *(const float4*)&A[j * RS + c4 * 4];
            float4 c = *(const float4*)&M2[m * RS + c4 * 4];
            res = make_float4(x.x + c.x, x.y + c.y, x.z + c.z, x.w + c.w);
        }
        gout[i] = res;
    }
}

extern "C" void kernel_launch(void* const* d_in, const int* in_sizes, int n_in,
                              void* d_out, int out_size, void* d_ws, size_t ws_size,
                              hipStream_t stream) {
    (void)n_in; (void)out_size; (void)d_ws; (void)ws_size;
    const float* m  = (const float*)d_in[0];
    const float* m1 = (const float*)d_in[1];
    const float* m2 = (const float*)d_in[2];
    float* out = (float*)d_out;

    const int B = in_sizes[0] / (TT * DD);   // 4096 batches
    const size_t shmem =
        (size_t)(3 * TT * RS + 2 * TT * 32 + 2 * TT + 2 * TT) * sizeof(float); // ~162 KB

    // CDNA5 WGP has 320KB LDS; raise the dynamic-LDS cap (idempotent, capture-safe).
    hipFuncSetAttribute(reinterpret_cast<const void*>(pfd_fused_kernel),
                        hipFuncAttributeMaxDynamicSharedMemorySize, (int)shmem);

    if (B > 0) {
        pfd_fused_kernel<<<dim3(B), dim3(BLK), shmem, stream>>>(m, m1, m2, out);
    }
}